// RAE_37254546325821
// MI455X (gfx1250) — compile-verified
//
#include <hip/hip_runtime.h>

// RAE LSTM autoencoder for MI455X (gfx1250, wave32).
// All heavy math on V_WMMA_F32_16X16X4_F32 (full fp32 matrix pipe);
// gate nonlinearities on the hardware V_TANH_F32 transcendental.
//
// Weights repacked 4-wide along k so each B-fragment is one coalesced
// global_load_b64:  element (k, g) of W^T lives at ((k>>2)*4H + g)*4 + (k&3).
// B-fragments are deliberately re-streamed from L2 every timestep (weights
// total ~4 MB << 192 MB L2); an opaque asm *offset* per timestep stops the
// compiler from hoisting them into (spilled) registers while preserving the
// global address space (global_load, not flat_load).
//
// Workspace (fp32 elements), total ~273 MB:
//   W4_e1[288*1024] W4_e2[384*512] W4_d1[256*512] W4_d2[384*1024] Wd4[256*32]
//   b_e1[1024] b_e2[512] b_d1[512] b_d2[1024]
//   h01[128*1024*256] (e1 out; reused as d2 out)
//   h1 [128*1024*128] (e2 out; t=T-1 slice = latent z)
//   h2 [128*1024*128] (d1 out)

typedef float v2f __attribute__((ext_vector_type(2)));
typedef float v8f __attribute__((ext_vector_type(8)));

#define WMMA_F32_16x16x4(a, b, c) \
  __builtin_amdgcn_wmma_f32_16x16x4_f32(false, (a), false, (b), (short)0, (c), false, false)

// Hardware tanh (gfx1250 V_TANH_F32); sigmoid(x) = 0.5*tanh(x/2) + 0.5.
#if __has_builtin(__builtin_amdgcn_tanhf)
__device__ __forceinline__ float htanh_(float x) { return __builtin_amdgcn_tanhf(x); }
#else
__device__ __forceinline__ float htanh_(float x) { return tanhf(x); }
#endif
__device__ __forceinline__ float hsig_(float x) { return __builtin_fmaf(0.5f, htanh_(0.5f * x), 0.5f); }

// ---------------------------------------------------------------------------
// Pack [Wih(4H,IN); Whh(4H,H)] -> 4-wide k-major W4, bias = bih + bhh.
// ---------------------------------------------------------------------------
__global__ void pack_lstm_weights(const float* __restrict__ Wih, const float* __restrict__ Whh,
                                  const float* __restrict__ bih, const float* __restrict__ bhh,
                                  float* __restrict__ W4, float* __restrict__ bias,
                                  int IN, int H) {
  const int fourH = 4 * H;
  const int K = IN + H;
  const int total = K * fourH;
  const int stride = gridDim.x * blockDim.x;
  for (int idx = blockIdx.x * blockDim.x + threadIdx.x; idx < total; idx += stride) {
    int quad = idx >> 2;
    int sub  = idx & 3;
    int g    = quad % fourH;
    int k    = (quad / fourH) * 4 + sub;
    W4[idx] = (k < IN) ? Wih[g * IN + k] : Whh[g * H + (k - IN)];
  }
  for (int idx = blockIdx.x * blockDim.x + threadIdx.x; idx < fourH; idx += stride)
    bias[idx] = bih[idx] + bhh[idx];
}

// Wd[N][K] -> 4-wide Wd4: element (k,f) at ((k>>2)*N + f)*4 + (k&3)
__global__ void pack_dense_weights(const float* __restrict__ Wd, float* __restrict__ Wd4,
                                   int K, int N) {
  const int stride = gridDim.x * blockDim.x;
  for (int idx = blockIdx.x * blockDim.x + threadIdx.x; idx < K * N; idx += stride) {
    int quad = idx >> 2;
    int sub  = idx & 3;
    int f    = quad % N;
    int k    = (quad / N) * 4 + sub;
    Wd4[idx] = Wd[f * K + k];
  }
}

// ---------------------------------------------------------------------------
// Persistent LSTM scan. grid.x = B/16; blockDim = 2*H (H/16 waves).
// Wave w owns H-columns [16w,16w+16) of all four gates (i,f,g,o), so the
// cell update is wave-local in the WMMA C-fragment registers.
// A-operand [x_t | h_{t-1}] lives in LDS; K = IN + H.
// ---------------------------------------------------------------------------
template <int IN, int H>
__global__ __launch_bounds__(2 * H) void lstm_scan(
    const float* __restrict__ x, long x_bstride, long x_tstride,
    const float* __restrict__ W4, const float* __restrict__ bias,
    float* __restrict__ hout, int T) {
  constexpr int K     = IN + H;
  constexpr int Kpad  = K + 4;            // multiple of 4 -> 8B-aligned v2f rows
  constexpr int fourH = 4 * H;
  constexpr int nthr  = 2 * H;

  __shared__ float A[16 * Kpad];

  const int tid  = threadIdx.x;
  const int lane = tid & 31;
  const int wave = tid >> 5;              // 0 .. H/16-1
  const int m16  = lane & 15;
  const int hi   = lane >> 4;             // 0: rows k0,k0+1 ; 1: rows k0+2,k0+3
  const int b0   = blockIdx.x * 16;
  const int ncol = wave * 16 + m16;       // this lane's column within H

  // Per-lane bases (hoisted): B-fragment, h-output, LDS h-write, LDS A-read.
  const float* wlane = W4 + (size_t)ncol * 4 + hi * 2;          // + k0*fourH + g*H*4
  float* hlane = hout + ((size_t)(b0 + hi * 8) * T) * H + ncol; // + j*T*H + t*H
  const int arow  = m16 * Kpad + hi * 2;                        // + k0 (LDS A fragment)
  const int alane = (hi * 8) * Kpad + IN + ncol;                // + j*Kpad (LDS h write)

  // Zero LDS (h-region must start as h_{-1} = 0).
  for (int idx = tid; idx < 16 * Kpad; idx += nthr) A[idx] = 0.0f;

  float gb[4];
#pragma unroll
  for (int g = 0; g < 4; ++g) gb[g] = bias[g * H + ncol];

  v8f c;
#pragma unroll
  for (int j = 0; j < 8; ++j) c[j] = 0.0f;

  __syncthreads();

  long tH = 0;  // t*H running offset for output stores
  for (int t = 0; t < T; ++t) {
    // Opaque *offset* (actually 0): stops LICM from hoisting the invariant
    // weight loads out of the 1024-step loop (which spilled to scratch),
    // while keeping pointer provenance -> global_load_b64, not flat_load.
    size_t woff = 0;
    asm volatile("" : "+v"(woff));
    const float* wIter = wlane + woff;

    // Stage x_t (coalesced; IN is a power of two -> shifts, unrolled trip count).
    {
      const float* xt = x + (long)t * x_tstride;
#pragma unroll
      for (int idx = tid; idx < 16 * IN; idx += nthr) {
        int m = idx / IN;
        int k = idx - m * IN;
        A[m * Kpad + k] = xt[(long)(b0 + m) * x_bstride + k];
      }
    }
    __syncthreads();  // x_t staged AND h_{t-1} (all waves) visible

    v8f acc[4];
#pragma unroll
    for (int g = 0; g < 4; ++g)
#pragma unroll
      for (int j = 0; j < 8; ++j) acc[g][j] = gb[g];

    // gates[16 x 4H] += A[16 x K] * W[K x 4H] (this wave's four 16-col slices).
    // One ds_load_b64 (A) + four global_load_b64 (B) + four WMMAs per k-step.
    // Bounded unroll keeps a rolling window of in-flight loads (no reg blowup).
#pragma unroll 4
    for (int k0 = 0; k0 < K; k0 += 4) {
      v2f a = *(const v2f*)(A + arow + k0);
      const float* wp = wIter + (size_t)k0 * fourH;  // == (k0>>2)*fourH*4
#pragma unroll
      for (int g = 0; g < 4; ++g) {
        v2f b = *(const v2f*)(wp + g * H * 4);
        acc[g] = WMMA_F32_16x16x4(a, b, acc[g]);
      }
    }
    __syncthreads();  // all waves done reading A before h/x regions are rewritten

    // Cell update + emit h_t (branch-free, hw V_TANH_F32).
    // C-fragment element j: (m = j + 8*hi, n = ncol).
#pragma unroll
    for (int j = 0; j < 8; ++j) {
      float iv = hsig_(acc[0][j]);
      float fv = hsig_(acc[1][j]);
      float gv = htanh_(acc[2][j]);
      float ov = hsig_(acc[3][j]);
      float cv = __builtin_fmaf(fv, c[j], iv * gv);
      c[j] = cv;
      float hv = ov * htanh_(cv);
      A[alane + j * Kpad] = hv;                    // next step's A operand
      hlane[(size_t)j * T * H + tH] = hv;          // layer output [B,T,H]
    }
    tH += H;
  }
}

// ---------------------------------------------------------------------------
// Dense head: out[rows x NO] = h3[rows x KD] * WdT[KD x NO] + bd.
// One wave per 16-row tile; NO/16 n-tiles; grid covers rows exactly.
// ---------------------------------------------------------------------------
template <int KD, int NO>
__global__ __launch_bounds__(256) void dense_proj(
    const float* __restrict__ h3, const float* __restrict__ Wd4,
    const float* __restrict__ bd, float* __restrict__ out, int rows) {
  const int lane = threadIdx.x & 31;
  const int wavesPerBlock = blockDim.x >> 5;
  const int tile = blockIdx.x * wavesPerBlock + (threadIdx.x >> 5);
  const int m16 = lane & 15;
  const int hi  = lane >> 4;
  const int row0 = tile * 16;
  if (row0 >= rows) return;  // wave-uniform

  constexpr int NT = NO / 16;
  v8f acc[NT];
#pragma unroll
  for (int nt = 0; nt < NT; ++nt) {
    float bv = bd[nt * 16 + m16];
#pragma unroll
    for (int j = 0; j < 8; ++j) acc[nt][j] = bv;
  }

  const float* arow  = h3 + (size_t)(row0 + m16) * KD + hi * 2;
  const float* wlane = Wd4 + (size_t)m16 * 4 + hi * 2;  // + k0*NO + nt*64
#pragma unroll 8
  for (int k0 = 0; k0 < KD; k0 += 4) {
    v2f a = *(const v2f*)(arow + k0);  // sequential 8B/lane: full line reuse
    const float* wp = wlane + (size_t)k0 * NO;
#pragma unroll
    for (int nt = 0; nt < NT; ++nt) {
      v2f b = *(const v2f*)(wp + nt * 64);
      acc[nt] = WMMA_F32_16x16x4(a, b, acc[nt]);
    }
  }

  float* olane = out + (size_t)(row0 + hi * 8) * NO + m16;
#pragma unroll
  for (int nt = 0; nt < NT; ++nt)
#pragma unroll
    for (int j = 0; j < 8; ++j) olane[j * NO + nt * 16] = acc[nt][j];
}

// ---------------------------------------------------------------------------
extern "C" void kernel_launch(void* const* d_in, const int* in_sizes, int n_in,
                              void* d_out, int out_size, void* d_ws, size_t ws_size,
                              hipStream_t stream) {
  (void)in_sizes; (void)n_in; (void)out_size; (void)ws_size;

  constexpr int B = 128, T = 1024, F = 32, E = 128, HID = 256;

  const float* x  = (const float*)d_in[0];
  const float* Wd = (const float*)d_in[1];
  const float* bd = (const float*)d_in[2];
  const float* e1_Wih = (const float*)d_in[3];
  const float* e1_Whh = (const float*)d_in[4];
  const float* e1_bih = (const float*)d_in[5];
  const float* e1_bhh = (const float*)d_in[6];
  const float* e2_Wih = (const float*)d_in[7];
  const float* e2_Whh = (const float*)d_in[8];
  const float* e2_bih = (const float*)d_in[9];
  const float* e2_bhh = (const float*)d_in[10];
  const float* d1_Wih = (const float*)d_in[11];
  const float* d1_Whh = (const float*)d_in[12];
  const float* d1_bih = (const float*)d_in[13];
  const float* d1_bhh = (const float*)d_in[14];
  const float* d2_Wih = (const float*)d_in[15];
  const float* d2_Whh = (const float*)d_in[16];
  const float* d2_bih = (const float*)d_in[17];
  const float* d2_bhh = (const float*)d_in[18];

  float* ws = (float*)d_ws;
  size_t off = 0;
  float* W4_e1 = ws + off; off += (size_t)(F + HID) * 4 * HID;   // 288*1024
  float* W4_e2 = ws + off; off += (size_t)(HID + E) * 4 * E;     // 384*512
  float* W4_d1 = ws + off; off += (size_t)(E + E) * 4 * E;       // 256*512
  float* W4_d2 = ws + off; off += (size_t)(E + HID) * 4 * HID;   // 384*1024
  float* Wd4   = ws + off; off += (size_t)HID * F;               // 256*32
  float* b_e1  = ws + off; off += 4 * HID;
  float* b_e2  = ws + off; off += 4 * E;
  float* b_d1  = ws + off; off += 4 * E;
  float* b_d2  = ws + off; off += 4 * HID;
  float* h01   = ws + off; off += (size_t)B * T * HID;           // e1 out / d2 out
  float* h1    = ws + off; off += (size_t)B * T * E;             // e2 out
  float* h2    = ws + off; off += (size_t)B * T * E;             // d1 out
  float* h3    = h01;                                            // reuse after e2

  // --- weight packing (tiny, full-chip) ---
  pack_lstm_weights<<<256, 256, 0, stream>>>(e1_Wih, e1_Whh, e1_bih, e1_bhh, W4_e1, b_e1, F,   HID);
  pack_lstm_weights<<<256, 256, 0, stream>>>(e2_Wih, e2_Whh, e2_bih, e2_bhh, W4_e2, b_e2, HID, E);
  pack_lstm_weights<<<256, 256, 0, stream>>>(d1_Wih, d1_Whh, d1_bih, d1_bhh, W4_d1, b_d1, E,   E);
  pack_lstm_weights<<<256, 256, 0, stream>>>(d2_Wih, d2_Whh, d2_bih, d2_bhh, W4_d2, b_d2, E,   HID);
  pack_dense_weights<<<32, 256, 0, stream>>>(Wd, Wd4, HID, F);

  // --- recurrent scans (8 workgroups = B/16 tiles; waves = H/16) ---
  lstm_scan<F, HID><<<dim3(B / 16), dim3(2 * HID), 0, stream>>>(
      x, (long)T * F, (long)F, W4_e1, b_e1, h01, T);
  lstm_scan<HID, E><<<dim3(B / 16), dim3(2 * E), 0, stream>>>(
      h01, (long)T * HID, (long)HID, W4_e2, b_e2, h1, T);
  // d1: input = z repeated -> point at e2's last timestep with t-stride 0
  lstm_scan<E, E><<<dim3(B / 16), dim3(2 * E), 0, stream>>>(
      h1 + (size_t)(T - 1) * E, (long)T * E, (long)0, W4_d1, b_d1, h2, T);
  lstm_scan<E, HID><<<dim3(B / 16), dim3(2 * HID), 0, stream>>>(
      h2, (long)T * E, (long)E, W4_d2, b_d2, h3, T);

  // --- dense head: [B*T, 256] x [256, 32] + bd ---
  // 8192 tiles of 16 rows; 8 waves/block -> 1024 blocks, exact cover.
  dense_proj<HID, F><<<dim3((B * T / 16) / 8), dim3(256), 0, stream>>>(
      h3, Wd4, bd, (float*)d_out, B * T);
}